// EvolvingGraphConv_32727650795618
// MI455X (gfx1250) — compile-verified
//
#include <hip/hip_runtime.h>
#include <hip/hip_bf16.h>

typedef __attribute__((ext_vector_type(2))) float v2f;
typedef __attribute__((ext_vector_type(8))) float v8f;

#define BB 4
#define TT 12
#define NN 1000
#define FF 64
#define SS 4096
#define S3 12288

__device__ __forceinline__ v8f wmma4(v2f a, v2f b, v8f c) {
    // D = A(16x4) * B(4x16) + C(16x16), all f32
    return __builtin_amdgcn_wmma_f32_16x16x4_f32(
        false, a, false, b, (short)0, c, false, false);
}

// ---------------- h init: h[b][k] = h0[k] ----------------
__global__ void init_k(const float* __restrict__ h0, float* __restrict__ h) {
    int idx = blockIdx.x * blockDim.x + threadIdx.x;   // 0..16383
    h[idx] = h0[idx & (SS - 1)];
}

// ---------------- summary[b][f] = mean_n x[b,t,n,f] ----------------
__global__ void summary_k(const float* __restrict__ x, int t,
                          float* __restrict__ summary) {
    int b = blockIdx.x, f = threadIdx.x;               // 4 blocks x 64
    const float* xb = x + (size_t)(b * TT + t) * NN * FF;
    float acc = 0.f;
    for (int n = 0; n < NN; ++n) acc += xb[n * FF + f];
    summary[b * FF + f] = acc * (1.0f / NN);
}

// ---------------- gi[b][j] = summary[b]·w_ih[j] + b_ih[j] ----------------
__global__ void gi_k(const float* __restrict__ summary,
                     const float* __restrict__ w_ih,
                     const float* __restrict__ b_ih,
                     float* __restrict__ gi) {
    int j = blockIdx.x * blockDim.x + threadIdx.x;     // 0..12287
    const float* wr = w_ih + (size_t)j * FF;
    float a0 = 0.f, a1 = 0.f, a2 = 0.f, a3 = 0.f;
    for (int f = 0; f < FF; f += 4) {
        float4 wv = *(const float4*)(wr + f);
        float4 s0 = *(const float4*)(summary + 0 * FF + f);
        float4 s1 = *(const float4*)(summary + 1 * FF + f);
        float4 s2 = *(const float4*)(summary + 2 * FF + f);
        float4 s3 = *(const float4*)(summary + 3 * FF + f);
        a0 += wv.x * s0.x + wv.y * s0.y + wv.z * s0.z + wv.w * s0.w;
        a1 += wv.x * s1.x + wv.y * s1.y + wv.z * s1.z + wv.w * s1.w;
        a2 += wv.x * s2.x + wv.y * s2.y + wv.z * s2.z + wv.w * s2.w;
        a3 += wv.x * s3.x + wv.y * s3.y + wv.z * s3.z + wv.w * s3.w;
    }
    float bb = b_ih[j];
    gi[0 * S3 + j] = a0 + bb;
    gi[1 * S3 + j] = a1 + bb;
    gi[2 * S3 + j] = a2 + bb;
    gi[3 * S3 + j] = a3 + bb;
}

// ---------------- gh[b][j] = h[b]·w_hh[j] + b_hh[j]  (streams 201MB) ------
// One wave per row j; lanes load contiguous float4 -> 512B/wave/iter.
__global__ void gh_k(const float* __restrict__ h,
                     const float* __restrict__ w_hh,
                     const float* __restrict__ b_hh,
                     float* __restrict__ gh) {
    int wave = threadIdx.x >> 5;
    int lane = threadIdx.x & 31;
    int j = blockIdx.x * 8 + wave;                     // 1536 blocks * 8 waves
    const float* wr = w_hh + (size_t)j * SS;
    float a0 = 0.f, a1 = 0.f, a2 = 0.f, a3 = 0.f;
    for (int k = lane * 4; k < SS; k += 128) {
        __builtin_prefetch(wr + k + 1024, 0, 1);       // stream w_hh ahead
        float4 wv = *(const float4*)(wr + k);
        float4 h0 = *(const float4*)(h + 0 * SS + k);
        float4 h1 = *(const float4*)(h + 1 * SS + k);
        float4 h2 = *(const float4*)(h + 2 * SS + k);
        float4 h3 = *(const float4*)(h + 3 * SS + k);
        a0 += wv.x * h0.x + wv.y * h0.y + wv.z * h0.z + wv.w * h0.w;
        a1 += wv.x * h1.x + wv.y * h1.y + wv.z * h1.z + wv.w * h1.w;
        a2 += wv.x * h2.x + wv.y * h2.y + wv.z * h2.z + wv.w * h2.w;
        a3 += wv.x * h3.x + wv.y * h3.y + wv.z * h3.z + wv.w * h3.w;
    }
    for (int off = 16; off; off >>= 1) {
        a0 += __shfl_down(a0, off, 32);
        a1 += __shfl_down(a1, off, 32);
        a2 += __shfl_down(a2, off, 32);
        a3 += __shfl_down(a3, off, 32);
    }
    if (lane == 0) {
        float bb = b_hh[j];
        gh[0 * S3 + j] = a0 + bb;
        gh[1 * S3 + j] = a1 + bb;
        gh[2 * S3 + j] = a2 + bb;
        gh[3 * S3 + j] = a3 + bb;
    }
}

// ---------------- GRU gates -> h_out (+ pair-packed weight copy) ----------
// wP[b][(f>>1)*128 + o*2 + (f&1)] = h_out[b][f*64+o]  (WMMA B-operand layout)
__global__ void gate_k(const float* __restrict__ gi, const float* __restrict__ gh,
                       const float* __restrict__ hin, float* __restrict__ hout,
                       float* __restrict__ wP) {
    int idx = blockIdx.x * blockDim.x + threadIdx.x;   // 0..16383
    int b = idx >> 12;
    int u = idx & (SS - 1);
    const float* gib = gi + b * S3;
    const float* ghb = gh + b * S3;
    float r = 1.f / (1.f + __expf(-(gib[u] + ghb[u])));
    float z = 1.f / (1.f + __expf(-(gib[u + SS] + ghb[u + SS])));
    float nn = tanhf(gib[u + 2 * SS] + r * ghb[u + 2 * SS]);
    float hv = hin[b * SS + u];
    float ho = (1.f - z) * nn + z * hv;
    hout[b * SS + u] = ho;
    int f = u >> 6, o = u & 63;
    wP[b * SS + (f >> 1) * 128 + o * 2 + (f & 1)] = ho;
}

// ---------------- inv[b][i] = rsqrt(clip(1 + sum_m adj[b,t,i,m])) --------
__global__ void deg_k(const float* __restrict__ adj, int t,
                      float* __restrict__ inv) {
    int wave = threadIdx.x >> 5;
    int lane = threadIdx.x & 31;
    int rowg = blockIdx.x * 8 + wave;                  // 500 blocks * 8 = 4000
    int b = rowg / NN, i = rowg % NN;
    const float* ar = adj + ((size_t)(b * TT + t) * NN + i) * NN;
    float s = 0.f;
    for (int m = lane; m < NN; m += 32) s += ar[m];
    for (int off = 16; off; off >>= 1) s += __shfl_down(s, off, 32);
    if (lane == 0) inv[rowg] = rsqrtf(fmaxf(s + 1.0f, 1e-8f));
}

// ---------------- ssup[b][m][o] = inv[m] * (xt[b] @ h.reshape(64,64)) -----
// WMMA f32 16x16x4: one wave computes 16 rows x 64 cols.
// Writes row-major ssup (diag term) AND pair-packed ssupP (B operand).
__global__ void sup_k(const float* __restrict__ x, const float* __restrict__ wP,
                      const float* __restrict__ inv, int t,
                      float* __restrict__ ssup, float* __restrict__ ssupP) {
    int wave = threadIdx.x >> 5, lane = threadIdx.x & 31;
    int mtile = blockIdx.x * 4 + wave;                 // grid.x = 16 -> 64 tiles
    if (mtile >= 63) return;                           // 63 tiles of 16 rows
    int b = blockIdx.y;
    int r = lane & 15, kh = lane >> 4;
    int row0 = mtile * 16;
    int row = row0 + r;
    int rowc = row < NN ? row : NN - 1;                // branch-free clamp
    const float* xb = x + (size_t)(b * TT + t) * NN * FF + (size_t)rowc * FF;
    const float* wb = wP + b * SS;                     // pair-packed 64x64 B
    v8f acc0 = {}, acc1 = {}, acc2 = {}, acc3 = {};
    for (int kb = 0; kb < FF; kb += 4) {
        int kk = kb + kh * 2;
        v2f a = *(const v2f*)(xb + kk);
        int pi = (kk >> 1) * 128 + r * 2;
        v2f b0 = *(const v2f*)(wb + pi +  0);
        v2f b1 = *(const v2f*)(wb + pi + 32);
        v2f b2 = *(const v2f*)(wb + pi + 64);
        v2f b3 = *(const v2f*)(wb + pi + 96);
        acc0 = wmma4(a, b0, acc0);
        acc1 = wmma4(a, b1, acc1);
        acc2 = wmma4(a, b2, acc2);
        acc3 = wmma4(a, b3, acc3);
    }
    const float* invb = inv + b * NN;
    float* sb = ssup + (size_t)b * NN * FF;
    float* sp = ssupP + (size_t)b * NN * FF;
    for (int i = 0; i < 8; ++i) {
        int m = row0 + i + kh * 8;
        if (m < NN) {
            float s = invb[m];
            float v0 = acc0[i] * s, v1 = acc1[i] * s;
            float v2 = acc2[i] * s, v3 = acc3[i] * s;
            sb[m * FF + r +  0] = v0;
            sb[m * FF + r + 16] = v1;
            sb[m * FF + r + 32] = v2;
            sb[m * FF + r + 48] = v3;
            int pb = (m >> 1) * 128 + (m & 1);
            sp[pb + (r +  0) * 2] = v0;
            sp[pb + (r + 16) * 2] = v1;
            sp[pb + (r + 32) * 2] = v2;
            sp[pb + (r + 48) * 2] = v3;
        }
    }
}

// ---------------- out = relu(inv[i]*(adj@ssup + ssup[i,:]) + bias) --------
__global__ void out_k(const float* __restrict__ adj, const float* __restrict__ ssup,
                      const float* __restrict__ ssupP,
                      const float* __restrict__ inv, const float* __restrict__ bias,
                      int t, float* __restrict__ out) {
    int wave = threadIdx.x >> 5, lane = threadIdx.x & 31;
    int mtile = blockIdx.x * 4 + wave;
    if (mtile >= 63) return;
    int b = blockIdx.y;
    int r = lane & 15, kh = lane >> 4;
    int row0 = mtile * 16;
    int row = row0 + r;
    int rowc = row < NN ? row : NN - 1;                // branch-free clamp
    const float* ab = adj + (size_t)(b * TT + t) * NN * NN + (size_t)rowc * NN;
    const float* sp = ssupP + (size_t)b * NN * FF;     // pair-packed B panel
    v8f acc0 = {}, acc1 = {}, acc2 = {}, acc3 = {};
    for (int kb = 0; kb < NN; kb += 4) {               // 1000 = 4*250 exact
        int kk = kb + kh * 2;
        v2f a = *(const v2f*)(ab + kk);
        int pi = (kk >> 1) * 128 + r * 2;
        v2f b0 = *(const v2f*)(sp + pi +  0);
        v2f b1 = *(const v2f*)(sp + pi + 32);
        v2f b2 = *(const v2f*)(sp + pi + 64);
        v2f b3 = *(const v2f*)(sp + pi + 96);
        acc0 = wmma4(a, b0, acc0);
        acc1 = wmma4(a, b1, acc1);
        acc2 = wmma4(a, b2, acc2);
        acc3 = wmma4(a, b3, acc3);
    }
    const float* invb = inv + b * NN;
    const float* sb = ssup + (size_t)b * NN * FF;
    float* ob = out + (size_t)(b * TT + t) * NN * FF;
    for (int i = 0; i < 8; ++i) {
        int m = row0 + i + kh * 8;
        if (m < NN) {
            float si = invb[m];
            float y0 = si * (acc0[i] + sb[m * FF + r +  0]) + bias[r +  0];
            float y1 = si * (acc1[i] + sb[m * FF + r + 16]) + bias[r + 16];
            float y2 = si * (acc2[i] + sb[m * FF + r + 32]) + bias[r + 32];
            float y3 = si * (acc3[i] + sb[m * FF + r + 48]) + bias[r + 48];
            ob[m * FF + r +  0] = fmaxf(y0, 0.f);
            ob[m * FF + r + 16] = fmaxf(y1, 0.f);
            ob[m * FF + r + 32] = fmaxf(y2, 0.f);
            ob[m * FF + r + 48] = fmaxf(y3, 0.f);
        }
    }
}

extern "C" void kernel_launch(void* const* d_in, const int* in_sizes, int n_in,
                              void* d_out, int out_size, void* d_ws, size_t ws_size,
                              hipStream_t stream) {
    const float* x    = (const float*)d_in[0];
    const float* adj  = (const float*)d_in[1];
    const float* w_ih = (const float*)d_in[2];
    const float* w_hh = (const float*)d_in[3];
    const float* b_ih = (const float*)d_in[4];
    const float* b_hh = (const float*)d_in[5];
    const float* h0   = (const float*)d_in[6];
    const float* bias = (const float*)d_in[7];
    float* out = (float*)d_out;

    float* ws      = (float*)d_ws;
    float* summary = ws;                 // 256
    float* gi      = summary + 256;      // 4*12288
    float* gh      = gi + 4 * S3;        // 4*12288
    float* hbuf0   = gh + 4 * S3;        // 4*4096
    float* hbuf1   = hbuf0 + 4 * SS;     // 4*4096
    float* wP      = hbuf1 + 4 * SS;     // 4*4096  pair-packed GRU weight
    float* inv     = wP + 4 * SS;        // 4*1000
    float* ssup    = inv + 4 * NN;       // 4*1000*64 row-major
    float* ssupP   = ssup + 4 * NN * FF; // 4*1000*64 pair-packed

    init_k<<<64, 256, 0, stream>>>(h0, hbuf0);
    for (int t = 0; t < TT; ++t) {
        float* hin  = (t & 1) ? hbuf1 : hbuf0;
        float* hout = (t & 1) ? hbuf0 : hbuf1;
        summary_k<<<BB, 64, 0, stream>>>(x, t, summary);
        gi_k<<<S3 / 256, 256, 0, stream>>>(summary, w_ih, b_ih, gi);
        gh_k<<<S3 / 8, 256, 0, stream>>>(hin, w_hh, b_hh, gh);
        gate_k<<<64, 256, 0, stream>>>(gi, gh, hin, hout, wP);
        deg_k<<<500, 256, 0, stream>>>(adj, t, inv);
        sup_k<<<dim3(16, BB), 128, 0, stream>>>(x, wP, inv, t, ssup, ssupP);
        out_k<<<dim3(16, BB), 128, 0, stream>>>(adj, ssup, ssupP, inv, bias, t, out);
    }
}